// Sampler_9904194584752
// MI455X (gfx1250) — compile-verified
//
#include <hip/hip_runtime.h>
#include <stdint.h>

#define THREADS  256
#define NBINS    4096
#define CAND_MAX 2048
#define KMAX     64
#define TILE     2048   // floats per async stage tile (8 KB)

// ---- CDNA5 async global->LDS path (inline asm; device pass only) --------
#if defined(__gfx1250__)
#  define HAVE_ASYNC_LDS 1
#else
#  define HAVE_ASYNC_LDS 0
#endif

__device__ __forceinline__ unsigned ordkey(float f) {
  unsigned u = __float_as_uint(f);
  return (u & 0x80000000u) ? ~u : (u | 0x80000000u);
}
__device__ __forceinline__ float ordinv(unsigned u) {
  return (u & 0x80000000u) ? __uint_as_float(u & 0x7fffffffu)
                           : __uint_as_float(~u);
}

#if HAVE_ASYNC_LDS
__device__ __forceinline__ void async_copy_b128(const float* g, float* l) {
  // Flat LDS address low 32 bits == raw LDS byte offset (ISA 10.2 aperture).
  unsigned lds_off = (unsigned)(uintptr_t)l;
  unsigned long long gaddr = (unsigned long long)(uintptr_t)g;
  // GV form: global_load_async_to_lds_b128 vdst(lds-addr), vaddr64, off
  asm volatile("global_load_async_to_lds_b128 %0, %1, off"
               :
               : "v"(lds_off), "v"(gaddr)
               : "memory");
}
__device__ __forceinline__ void wait_async_zero() {
#  if __has_builtin(__builtin_amdgcn_s_wait_asynccnt)
  __builtin_amdgcn_s_wait_asynccnt(0);
#  else
  asm volatile("s_wait_asynccnt 0" ::: "memory");
#  endif
}
#endif

__global__ __launch_bounds__(THREADS)
void sampler_kernel(const float* __restrict__ logits,
                    const float* __restrict__ temperature,
                    const float* __restrict__ top_p,
                    const int*   __restrict__ token_lengths,
                    const int*   __restrict__ top_k_ptr,
                    int*         __restrict__ out,
                    int B, int V)
{
  __shared__ unsigned hist[NBINS];
  __shared__ unsigned chunk[THREADS];
  __shared__ unsigned cand_u[CAND_MAX];
  __shared__ int      cand_i[CAND_MAX];
  __shared__ unsigned long long red[THREADS];
  __shared__ unsigned top_u[KMAX];
  __shared__ int      top_i[KMAX];
  __shared__ int sh_cnt;
  __shared__ int sh_thrbin;
#if HAVE_ASYNC_LDS
  __shared__ __align__(16) float stage[2][TILE];
#endif

  const int row = blockIdx.x;
  if (row >= B) return;
  const int tid = threadIdx.x;

#if defined(__HIP_DEVICE_COMPILE__)
#  if __has_builtin(__builtin_amdgcn_s_wait_tensorcnt)
  __builtin_amdgcn_s_wait_tensorcnt(0);   // no-op marker; TENSORcnt==0
#  endif
#endif

  int k = *top_k_ptr;
  if (k < 1) k = 1;
  if (k > KMAX) k = KMAX;
  if (k > V) k = V;
  const float* rowp = logits + (size_t)row * (size_t)V;

  for (int i = tid; i < NBINS; i += THREADS) hist[i] = 0u;
  if (tid == 0) sh_cnt = 0;
  __syncthreads();

  // ---------------- Pass A: 12-bit ordered-key histogram ----------------
#if HAVE_ASYNC_LDS
  const int nt = V / TILE;
  if (nt > 0) {
    for (int s = tid; s < TILE / 4; s += THREADS)
      async_copy_b128(rowp + s * 4, &stage[0][s * 4]);
    for (int t = 0; t < nt; ++t) {
      wait_async_zero();
      __syncthreads();                    // stage[t&1] ready; prev consume done
      if (t + 1 < nt) {
        const float* src = rowp + (size_t)(t + 1) * TILE;
        float* dst = &stage[(t + 1) & 1][0];
        for (int s = tid; s < TILE / 4; s += THREADS)
          async_copy_b128(src + s * 4, dst + s * 4);   // overlaps with binning
      }
      const float* buf = &stage[t & 1][0];
      for (int i = tid; i < TILE; i += THREADS) {
        unsigned u = ordkey(buf[i]);
        atomicAdd(&hist[u >> 20], 1u);
      }
    }
  }
  for (int j = nt * TILE + tid; j < V; j += THREADS) {
    unsigned u = ordkey(rowp[j]);
    atomicAdd(&hist[u >> 20], 1u);
  }
#else
  for (int j = tid; j < V; j += THREADS) {
    __builtin_prefetch(rowp + j + 16 * THREADS, 0, 1);  // global_prefetch_b8
    unsigned u = ordkey(rowp[j]);
    atomicAdd(&hist[u >> 20], 1u);
  }
#endif
  __syncthreads();

  // ------------- threshold bin: largest b with suffix-count >= k --------
  {
    const int CB = NBINS / THREADS;       // 16
    unsigned s0 = 0;
    const int base = tid * CB;
    for (int i = 0; i < CB; ++i) s0 += hist[base + i];
    chunk[tid] = s0;
  }
  __syncthreads();
  if (tid == 0) {
    const int CB = NBINS / THREADS;
    int cum = 0, b = 0, c = THREADS - 1;
    for (; c >= 0; --c) {
      if (cum + (int)chunk[c] >= k) break;
      cum += (int)chunk[c];
    }
    if (c >= 0) {
      int bb = c * CB + CB - 1;
      for (; bb >= c * CB; --bb) {
        cum += (int)hist[bb];
        if (cum >= k) break;
      }
      b = (bb < c * CB) ? c * CB : bb;
    }
    sh_thrbin = b;
  }
  __syncthreads();

  // ---------------- Pass B: gather candidates (L2-resident) --------------
  const unsigned thrbin = (unsigned)sh_thrbin;
  for (int j = tid; j < V; j += THREADS) {
    __builtin_prefetch(rowp + j + 16 * THREADS, 0, 2);
    unsigned u = ordkey(rowp[j]);
    if ((u >> 20) >= thrbin) {
      int pos = atomicAdd(&sh_cnt, 1);
      if (pos < CAND_MAX) { cand_u[pos] = u; cand_i[pos] = j; }
    }
  }
  __syncthreads();
  const int n = (sh_cnt < CAND_MAX) ? sh_cnt : CAND_MAX;

  // ------- Phase C: k iterations of block argmax (desc val, asc idx) -----
  for (int r = 0; r < k; ++r) {
    unsigned long long best = 0ull;
    for (int c = tid; c < n; c += THREADS) {
      unsigned u = cand_u[c];
      if (u) {
        unsigned long long key =
            ((unsigned long long)u << 32) | (unsigned)(~(unsigned)cand_i[c]);
        if (key > best) best = key;
      }
    }
    red[tid] = best;
    __syncthreads();
    for (int s = THREADS / 2; s > 0; s >>= 1) {
      if (tid < s && red[tid + s] > red[tid]) red[tid] = red[tid + s];
      __syncthreads();
    }
    const unsigned long long win = red[0];
    if (win != 0ull) {
      const unsigned wu = (unsigned)(win >> 32);
      const int wi = (int)(~(unsigned)(win & 0xffffffffull));
      if (tid == 0) { top_u[r] = wu; top_i[r] = wi; }
      for (int c = tid; c < n; c += THREADS)
        if (cand_u[c] == wu && cand_i[c] == wi) cand_u[c] = 0u;
    } else if (tid == 0) {
      top_u[r] = 0u; top_i[r] = 0;
    }
    __syncthreads();
  }

  // ---------------- Phase D: per-row epilogue (one lane) -----------------
  if (tid == 0) {
    const float t = temperature[row];
    const float p = top_p[row];
    float val[KMAX]; int idx[KMAX];
    for (int r = 0; r < k; ++r) {
      unsigned u = top_u[r];
      val[r] = (u != 0u) ? ordinv(u) : -__builtin_inff();
      idx[r] = top_i[r];
    }
    const int greedy = idx[0];

    // softmax over scaled top-k (all finite; max is slot 0)
    float e[KMAX];
    const float mx = val[0] / t;
    float Z = 0.f;
    for (int r = 0; r < k; ++r) {
      float s = val[r] / t;
      e[r] = expf(s - mx);
      Z += e[r];
    }
    // top-p: ascending cumsum (same accumulation order as reference);
    // keep slot r iff sum_{j>=r} q_j > 1-p ; slot 0 always kept
    int m;
    {
      int keepflag[KMAX];
      float acc = 0.f;
      for (int r = k - 1; r >= 0; --r) {
        acc += e[r] / Z;
        keepflag[r] = (acc > 1.0f - p) ? 1 : 0;
      }
      keepflag[0] = 1;
      m = 0;
      while (m < k && keepflag[m]) ++m;   // kept set is a prefix
    }
    // renormalized probs over kept prefix (second softmax of reference)
    float q2[KMAX];
    float Z2 = 0.f;
    for (int r = 0; r < m; ++r) Z2 += e[r];
    for (int r = 0; r < m; ++r) q2[r] = e[r] / Z2;
    for (int r = m; r < k; ++r) q2[r] = 0.f;

    // filler indices: top_k over -inf-scattered array picks smallest unused ids
    {
      int c = 0;
      for (int r = m; r < k; ++r) {
        for (;;) {
          bool used = false;
          for (int j2 = 0; j2 < m; ++j2)
            if (idx[j2] == c) { used = true; break; }
          if (!used) break;
          ++c;
        }
        idx[r] = (c < V) ? c : (V - 1);
        ++c;
      }
    }
    // length score softmax over all k slots (fillers included, as in ref)
    float lv[KMAX]; float lmax = -3.4e38f;
    for (int r = 0; r < k; ++r) {
      lv[r] = (float)token_lengths[idx[r]];
      lmax = fmaxf(lmax, lv[r]);
    }
    float Ls = 0.f;
    for (int r = 0; r < k; ++r) { lv[r] = expf(lv[r] - lmax); Ls += lv[r]; }

    float mix[KMAX];
    for (int r = 0; r < k; ++r) {
      float mr = 0.5f * q2[r] + 0.5f * (lv[r] / Ls);     // MIX_RATIO = 0.5
      mix[r] = (q2[r] >= 0.001f) ? mr : -__builtin_inff();
    }
    // EOS rule (EOS_TOKEN_ID=2, finite slots only)
    bool has_eos = false; float eprob = 0.f;
    for (int r = 0; r < m; ++r)
      if (idx[r] == 2) { has_eos = true; eprob += q2[r]; }
    if (has_eos) {
      float th2 = fmaxf(eprob * 0.01f, 0.005f);
      for (int r = 0; r < k; ++r)
        if (!(q2[r] >= th2)) mix[r] = -__builtin_inff();
    }
    // first-occurrence argmax (jnp.argmax semantics)
    int best = 0; float bm = mix[0];
    for (int r = 1; r < k; ++r)
      if (mix[r] > bm) { bm = mix[r]; best = r; }
    const int chosen = idx[best];
    out[row] = (t < 1e-5f) ? greedy : chosen;
  }
}

extern "C" void kernel_launch(void* const* d_in, const int* in_sizes, int n_in,
                              void* d_out, int out_size, void* d_ws, size_t ws_size,
                              hipStream_t stream) {
  (void)n_in; (void)out_size; (void)d_ws; (void)ws_size;
  const float* logits        = (const float*)d_in[0];
  const float* temperature   = (const float*)d_in[1];
  const float* top_p         = (const float*)d_in[2];
  const int*   token_lengths = (const int*)d_in[3];
  const int*   top_k         = (const int*)d_in[4];
  const int B = in_sizes[1];
  if (B <= 0) return;
  const int V = in_sizes[0] / B;
  int* out = (int*)d_out;
  sampler_kernel<<<dim3(B), dim3(THREADS), 0, stream>>>(
      logits, temperature, top_p, token_lengths, top_k, out, B, V);
}